// KeyConvMemoryMultiHead_4080218931761
// MI455X (gfx1250) — compile-verified
//
#include <hip/hip_runtime.h>

// Fused KeyConvMemoryMultiHead for gfx1250 (MI455X), fp32 WMMA 16x16x4.
// One workgroup = one (batch, row, 16-pixel) tile; 8 waves = 8 heads.

typedef __attribute__((ext_vector_type(2))) float v2f;
typedef __attribute__((ext_vector_type(8))) float v8f;

#define IN_C    256
#define NH      8
#define KD2     288      // KEY_DIM * 3 * 3
#define MEM     256
#define HD      64
#define OUTC    256
#define HW      64
#define L_IMG   4096

// LDS layout (dwords). Strides padded to avoid 64-bank conflicts.
#define LDS_X    0                    // Xhalo  [256][68]  (q = rr*18+cc, valid 0..53, zero 54..63)
#define LDS_P    17408                // Ptile  [256][68]
#define LDS_ATT  34816                // att    [8*256][18]
#define LDS_O    71680                // headO  [512][18]
#define LDS_DW   80896                // total dwords = 323584 bytes (<= 320KB WGP LDS)

static __device__ __forceinline__ v8f wmma4(v2f a, v2f b, v8f c) {
  // D = A(16x4 f32) * B(4x16 f32) + C(16x16 f32); 8-arg VOP3P form.
  return __builtin_amdgcn_wmma_f32_16x16x4_f32(
      /*neg_a=*/false, a, /*neg_b=*/false, b,
      /*c_mod=*/(short)0, c, /*reuse_a=*/false, /*reuse_b=*/false);
}

__global__ __launch_bounds__(256)
void kcmm_fused_gfx1250(const float* __restrict__ x,
                        const float* __restrict__ Wp,
                        const float* __restrict__ bp,
                        const float* __restrict__ keyb,   // [8][288][256]
                        const float* __restrict__ memb,   // [8][256][64]
                        const float* __restrict__ Wo,     // [256][512]
                        const float* __restrict__ bo,     // [256]
                        float* __restrict__ out)          // [8][256][4096]
{
  extern __shared__ float lds[];

  const int tid  = threadIdx.x;
  const int wave = tid >> 5;          // 0..7 == head id
  const int lane = tid & 31;          // wave32
  const int half = lane >> 4;         // lane/16
  const int li   = lane & 15;         // M/N index within fragment

  const int blk = blockIdx.x;         // 0..2047
  const int b   = blk >> 8;
  const int rem = blk & 255;
  const int y   = rem >> 2;
  const int x0  = (rem & 3) << 4;

  const v8f vzero = {0.f,0.f,0.f,0.f,0.f,0.f,0.f,0.f};

  // ---------------- Phase 0: load x halo (3 rows x 18 cols x 256 ch) ----------------
  for (int idx = tid; idx < IN_C * 54; idx += 256) {
    int c  = idx / 54;
    int r2 = idx - c * 54;
    int rr = r2 / 18;
    int cc = r2 - rr * 18;
    int gy = y - 1 + rr;
    int gx = x0 - 1 + cc;
    float v = 0.f;
    if (gy >= 0 && gy < HW && gx >= 0 && gx < HW)
      v = x[((b * IN_C + c) * HW + gy) * HW + gx];
    lds[LDS_X + c * 68 + rr * 18 + cc] = v;
  }
  for (int idx = tid; idx < IN_C * 14; idx += 256) {   // zero q = 54..67
    int c = idx / 14;
    int q = 54 + (idx - c * 14);
    lds[LDS_X + c * 68 + q] = 0.f;
  }
  __syncthreads();

  // ---------------- Phase 1: proj GEMM  Ptile[o][q] = Wp[o][:] . Xhalo[:][q] ---------
  {
    v8f acc[2][4];
    #pragma unroll
    for (int j = 0; j < 2; ++j)
      #pragma unroll
      for (int qt = 0; qt < 4; ++qt) acc[j][qt] = vzero;

    for (int k0 = 0; k0 < IN_C; k0 += 4) {
      const int kl = k0 + 2 * half;
      v2f bf[4];
      #pragma unroll
      for (int qt = 0; qt < 4; ++qt) {
        bf[qt].x = lds[LDS_X + kl * 68 + qt * 16 + li];
        bf[qt].y = lds[LDS_X + (kl + 1) * 68 + qt * 16 + li];
      }
      #pragma unroll
      for (int j = 0; j < 2; ++j) {
        const int o0 = (2 * wave + j) * 16;
        v2f af;
        af.x = Wp[(o0 + li) * IN_C + kl];
        af.y = Wp[(o0 + li) * IN_C + kl + 1];
        #pragma unroll
        for (int qt = 0; qt < 4; ++qt) acc[j][qt] = wmma4(af, bf[qt], acc[j][qt]);
      }
    }
    #pragma unroll
    for (int j = 0; j < 2; ++j)
      #pragma unroll
      for (int r = 0; r < 8; ++r) {
        const int o = (2 * wave + j) * 16 + 8 * half + r;
        const float bv = bp[o];
        #pragma unroll
        for (int qt = 0; qt < 4; ++qt)
          lds[LDS_P + o * 68 + qt * 16 + li] = acc[j][qt][r] + bv;
      }
  }
  __syncthreads();

  // ---------------- Phase 2: attention logits per head (wave == head) ---------------
  // att[i][x] = sum_k keyb[n][k][i] * q[k][x],  q[k][x] = Ptile[c][kh*18+kw + x]
  const int n = wave;
  v8f att[16];
  #pragma unroll
  for (int t = 0; t < 16; ++t) att[t] = vzero;

  for (int k0 = 0; k0 < KD2; k0 += 4) {
    const int kl = k0 + 2 * half;
    v2f bf;
    #pragma unroll
    for (int v = 0; v < 2; ++v) {
      const int ch = (kl + v) * NH + n;       // global unfold channel = c*9 + p
      const int c  = ch / 9;
      const int p  = ch - c * 9;
      const int kh = p / 3;
      const int kw = p - kh * 3;
      bf[v] = lds[LDS_P + c * 68 + kh * 18 + kw + li];
    }
    #pragma unroll
    for (int t = 0; t < 16; ++t) {
      v2f af;
      af.x = keyb[(n * KD2 + kl) * MEM + t * 16 + li];
      af.y = keyb[(n * KD2 + kl + 1) * MEM + t * 16 + li];
      att[t] = wmma4(af, bf, att[t]);
    }
  }

  // ---------------- softmax over i (256) for each column x = li ---------------------
  {
    float mx = -3.0e38f;
    #pragma unroll
    for (int t = 0; t < 16; ++t)
      #pragma unroll
      for (int r = 0; r < 8; ++r) mx = fmaxf(mx, att[t][r]);
    mx = fmaxf(mx, __shfl_xor(mx, 16, 32));   // combine lane halves (same column)

    float sum = 0.f;
    #pragma unroll
    for (int t = 0; t < 16; ++t)
      #pragma unroll
      for (int r = 0; r < 8; ++r) {
        float e = __expf(att[t][r] - mx);
        att[t][r] = e;
        sum += e;
      }
    sum += __shfl_xor(sum, 16, 32);
    const float inv = 1.f / sum;

    #pragma unroll
    for (int t = 0; t < 16; ++t)
      #pragma unroll
      for (int r = 0; r < 8; ++r) {
        const int i = t * 16 + 8 * half + r;
        lds[LDS_ATT + (n * MEM + i) * 18 + li] = att[t][r] * inv;
      }
  }

  // ---------------- Phase 3: memory readback  o[d][x] = memb[n][:,d] . att[:, x] ----
  {
    v8f oacc[4];
    #pragma unroll
    for (int t = 0; t < 4; ++t) oacc[t] = vzero;

    for (int k0 = 0; k0 < MEM; k0 += 4) {
      const int kl = k0 + 2 * half;
      v2f bf;
      bf.x = lds[LDS_ATT + (n * MEM + kl) * 18 + li];
      bf.y = lds[LDS_ATT + (n * MEM + kl + 1) * 18 + li];
      #pragma unroll
      for (int t = 0; t < 4; ++t) {
        v2f af;
        af.x = memb[(n * MEM + kl) * HD + t * 16 + li];
        af.y = memb[(n * MEM + kl + 1) * HD + t * 16 + li];
        oacc[t] = wmma4(af, bf, oacc[t]);
      }
    }
    #pragma unroll
    for (int t = 0; t < 4; ++t)
      #pragma unroll
      for (int r = 0; r < 8; ++r) {
        const int d = t * 16 + 8 * half + r;
        lds[LDS_O + (n * HD + d) * 18 + li] = oacc[t][r];
      }
  }
  __syncthreads();

  // ---------------- Phase 4: out projection  out[oo][x] = Wo[oo][:] . headO[:, x] ---
  {
    v8f facc[2];
    facc[0] = vzero; facc[1] = vzero;

    for (int k0 = 0; k0 < NH * HD; k0 += 4) {
      const int kl = k0 + 2 * half;
      v2f bf;
      bf.x = lds[LDS_O + kl * 18 + li];
      bf.y = lds[LDS_O + (kl + 1) * 18 + li];
      #pragma unroll
      for (int j = 0; j < 2; ++j) {
        const int o0 = (2 * wave + j) * 16;
        v2f af;
        af.x = Wo[(o0 + li) * (NH * HD) + kl];
        af.y = Wo[(o0 + li) * (NH * HD) + kl + 1];
        facc[j] = wmma4(af, bf, facc[j]);
      }
    }
    #pragma unroll
    for (int j = 0; j < 2; ++j)
      #pragma unroll
      for (int r = 0; r < 8; ++r) {
        const int oo = (2 * wave + j) * 16 + 8 * half + r;
        out[(b * OUTC + oo) * L_IMG + y * HW + x0 + li] = facc[j][r] + bo[oo];
      }
  }
}

extern "C" void kernel_launch(void* const* d_in, const int* in_sizes, int n_in,
                              void* d_out, int out_size, void* d_ws, size_t ws_size,
                              hipStream_t stream) {
  (void)in_sizes; (void)n_in; (void)d_ws; (void)ws_size; (void)out_size;
  const float* x    = (const float*)d_in[0];
  const float* Wp   = (const float*)d_in[1];
  const float* bpj  = (const float*)d_in[2];
  const float* keyb = (const float*)d_in[3];
  const float* memb = (const float*)d_in[4];
  const float* Wo   = (const float*)d_in[5];
  const float* bo   = (const float*)d_in[6];
  float* out = (float*)d_out;

  const size_t shmem = (size_t)LDS_DW * sizeof(float);   // 323,584 B (< 320KB WGP LDS)
  dim3 grid(8 * 64 * 4);   // batch * rows * 4 col-tiles
  dim3 block(256);         // 8 wave32 = 8 heads
  kcmm_fused_gfx1250<<<grid, block, shmem, stream>>>(x, Wp, bpj, keyb, memb, Wo, bo, out);
}